// MultiBoxLoss_12300786336362
// MI455X (gfx1250) — compile-verified
//
#include <hip/hip_runtime.h>
#include <hip/hip_bf16.h>

// ---------------- problem constants (match reference) ----------------
#define BATCH      16
#define NPRIORS    131072
#define NGT        32
#define TPB        256
#define BLK_P      (NPRIORS / TPB)      // 512 blocks per image
#define NB_TOTAL   (BATCH * BLK_P)      // 8192 block-partials per stream
#define NBINS      512
#define BIN_SCALE  64.0f                // bins cover CE in [0, 8)
#define OVTH       0.35f

// ---------------- workspace layout (bytes) ----------------
#define OFF_BEST   0                       // u64 best[B][G]      : 4096
#define OFF_NPOS   4096                    // int  npos[B]        : 64 (pad 128)
#define OFF_HCNT   4224                    // u32  hcnt[B][NBINS] : 32768
#define OFF_HSUM   36992                   // f32  hsum[B][NBINS] : 32768
#define OFF_PART   69760                   // f32  4 streams x 8192
#define PART_STRIDE (NB_TOTAL * 4)         // bytes per stream = 32768

typedef float v2f __attribute__((ext_vector_type(2)));
typedef float v8f __attribute__((ext_vector_type(8)));
typedef int   v4i __attribute__((vector_size(4 * sizeof(int))));

// Async Global->LDS datapath (gfx1250): gate on builtin availability so the
// source stays portable across ROCm 7.2 / amdgpu-toolchain header sets.
#if defined(__has_builtin)
#if __has_builtin(__builtin_amdgcn_global_load_async_to_lds_b128) && \
    __has_builtin(__builtin_amdgcn_s_wait_asynccnt)
#define HAVE_ASYNC_LDS 1
#endif
#endif
#ifndef HAVE_ASYNC_LDS
#define HAVE_ASYNC_LDS 0
#endif

__device__ __forceinline__ float sl1(float d) {
    d = fabsf(d);
    return d < 1.0f ? 0.5f * d * d : d - 0.5f;
}

// ---------------- kernel 0: init workspace ----------------
__global__ void init_ws(unsigned long long* __restrict__ best,
                        unsigned int* __restrict__ zwords, int nzero) {
    int i = blockIdx.x * blockDim.x + threadIdx.x;
    if (i < BATCH * NGT) best[i] = 0x00000000FFFFFFFFull;   // iou=0.0, p=~0 decoded
    if (i < nzero)       zwords[i] = 0u;
}

// ---------------- kernel 1: per-truth argmax over priors ----------------
// key = (float_bits(iou) << 32) | ~p  -> max key == max iou, ties -> smallest p
__global__ void best_prior_kernel(const float* __restrict__ priors,
                                  const float* __restrict__ targets,
                                  unsigned long long* __restrict__ best) {
    __shared__ float tr[NGT * 4];
    __shared__ float tarea[NGT];
    __shared__ unsigned long long lk[NGT];
    const int b   = blockIdx.y;
    const int p0  = blockIdx.x * TPB;
    const int tid = threadIdx.x;

    if (tid < NGT * 4) {
        int g = tid >> 2, j = tid & 3;
        tr[tid] = targets[((size_t)b * NGT + g) * 15 + j];
    }
    if (tid < NGT) lk[tid] = 0x00000000FFFFFFFFull;
    __syncthreads();
    if (tid < NGT)
        tarea[tid] = (tr[tid*4+2] - tr[tid*4+0]) * (tr[tid*4+3] - tr[tid*4+1]);
    __syncthreads();

    const int p = p0 + tid;
    const float4 pr = ((const float4*)priors)[p];
    const float bx0 = pr.x - pr.z * 0.5f, by0 = pr.y - pr.w * 0.5f;
    const float bx1 = pr.x + pr.z * 0.5f, by1 = pr.y + pr.w * 0.5f;
    const float areab = pr.z * pr.w;
    const unsigned int pinv = ~(unsigned int)p;

    for (int g = 0; g < NGT; ++g) {
        float iw = fminf(tr[g*4+2], bx1) - fmaxf(tr[g*4+0], bx0);
        float ih = fminf(tr[g*4+3], by1) - fmaxf(tr[g*4+1], by0);
        iw = fmaxf(iw, 0.0f); ih = fmaxf(ih, 0.0f);
        float inter = iw * ih;
        float iou = inter / (tarea[g] + areab - inter);
        unsigned long long key =
            ((unsigned long long)__float_as_uint(iou) << 32) | (unsigned long long)pinv;
        // wave32 max-reduce
        for (int m = 16; m; m >>= 1) {
            unsigned long long o = __shfl_xor(key, m, 32);
            key = (o > key) ? o : key;
        }
        if ((tid & 31) == 0) atomicMax(&lk[g], key);
    }
    __syncthreads();
    if (tid < NGT) atomicMax(&best[b * NGT + tid], lk[tid]);
}

// ---------------- kernel 2: fused match + losses + CE histogram ----------------
__global__ void main_kernel(const float* __restrict__ conf,
                            const float* __restrict__ loc,
                            const float* __restrict__ landm,
                            const float* __restrict__ priors,
                            const float* __restrict__ targets,
                            const unsigned long long* __restrict__ best,
                            int* __restrict__ npos_img,
                            unsigned int* __restrict__ hcnt,
                            float* __restrict__ hsum,
                            float* __restrict__ pbox, float* __restrict__ plandm,
                            float* __restrict__ pce,  float* __restrict__ pnpos) {
    __shared__ float tr[NGT * 15];
    __shared__ float tarea[NGT];
    __shared__ unsigned int bp[NGT];
    __shared__ float lm[TPB * 10];
    __shared__ unsigned int shcnt[NBINS];
    __shared__ float shsum[NBINS];
    __shared__ float4 red[TPB];

    const int b   = blockIdx.y;
    const int p0  = blockIdx.x * TPB;
    const int tid = threadIdx.x;

    // stage landm tile into LDS: async DMA path on CDNA5, float4 fallback else
    const float* lbase = landm + ((size_t)b * NPRIORS + p0) * 10;
    const float4* lsrc = (const float4*)lbase;
#if HAVE_ASYNC_LDS
    {
        typedef v4i __attribute__((address_space(1)))* gp4_t;   // global, pointee v4i
        typedef v4i __attribute__((address_space(3)))* lp4_t;   // LDS, pointee v4i
        for (int i = tid; i < TPB * 10 / 4; i += TPB)
            __builtin_amdgcn_global_load_async_to_lds_b128(
                (gp4_t)(lsrc + i),
                (lp4_t)((float4*)lm + i), 0, 0);
    }
#else
    for (int i = tid; i < TPB * 10 / 4; i += TPB) ((float4*)lm)[i] = lsrc[i];
#endif
    __builtin_prefetch(lbase + TPB * 10, 0, 1);   // next tile -> global_prefetch_b8

    for (int i = tid; i < NGT * 15; i += TPB) tr[i] = targets[(size_t)b * NGT * 15 + i];
    for (int i = tid; i < NBINS; i += TPB) { shcnt[i] = 0u; shsum[i] = 0.0f; }
    if (tid < NGT) bp[tid] = ~(unsigned int)(best[b * NGT + tid] & 0xFFFFFFFFull);
#if HAVE_ASYNC_LDS
    __builtin_amdgcn_s_wait_asynccnt(0);
#endif
    __syncthreads();
    if (tid < NGT)
        tarea[tid] = (tr[tid*15+2] - tr[tid*15+0]) * (tr[tid*15+3] - tr[tid*15+1]);
    __syncthreads();

    const int p = p0 + tid;
    const float4 pr = ((const float4*)priors)[p];
    const float bx0 = pr.x - pr.z * 0.5f, by0 = pr.y - pr.w * 0.5f;
    const float bx1 = pr.x + pr.z * 0.5f, by1 = pr.y + pr.w * 0.5f;
    const float areab = pr.z * pr.w;

    // argmax over truths (first max, like jnp.argmax)
    float bestov = -1.0f; int gi = 0;
    for (int g = 0; g < NGT; ++g) {
        float iw = fminf(tr[g*15+2], bx1) - fmaxf(tr[g*15+0], bx0);
        float ih = fminf(tr[g*15+3], by1) - fmaxf(tr[g*15+1], by0);
        iw = fmaxf(iw, 0.0f); ih = fmaxf(ih, 0.0f);
        float inter = iw * ih;
        float iou = inter / (tarea[g] + areab - inter);
        if (iou > bestov) { bestov = iou; gi = g; }
    }
    // force-match override (ascending g: last writer wins, matching .at[].set)
    for (int g = 0; g < NGT; ++g)
        if (bp[g] == (unsigned int)p) { bestov = 2.0f; gi = g; }

    const bool pos = bestov >= OVTH;   // labels are all 1

    // two-class CE with log-softmax
    const float2 cd = ((const float2*)conf)[(size_t)b * NPRIORS + p];
    const float mx  = fmaxf(cd.x, cd.y);
    const float lse = mx + logf(expf(cd.x - mx) + expf(cd.y - mx));
    const float ce  = lse - (pos ? cd.y : cd.x);

    float lb = 0.0f, ll = 0.0f, pce_v = 0.0f, posf = 0.0f;
    if (pos) {
        posf = 1.0f; pce_v = ce;
        const float tx0 = tr[gi*15+0], ty0 = tr[gi*15+1];
        const float tx1 = tr[gi*15+2], ty1 = tr[gi*15+3];
        const float ivx = 1.0f / (0.1f * pr.z), ivy = 1.0f / (0.1f * pr.w);
        const float gcx = ((tx0 + tx1) * 0.5f - pr.x) * ivx;
        const float gcy = ((ty0 + ty1) * 0.5f - pr.y) * ivy;
        const float gw  = logf((tx1 - tx0) / pr.z) * 5.0f;   // / 0.2
        const float gh  = logf((ty1 - ty0) / pr.w) * 5.0f;
        const float4 ld = ((const float4*)loc)[(size_t)b * NPRIORS + p];
        lb = sl1(ld.x - gcx) + sl1(ld.y - gcy) + sl1(ld.z - gw) + sl1(ld.w - gh);
        #pragma unroll
        for (int k = 0; k < 5; ++k) {
            const float gx = (tr[gi*15+4+2*k]   - pr.x) * ivx;
            const float gy = (tr[gi*15+4+2*k+1] - pr.y) * ivy;
            ll += sl1(lm[tid*10+2*k] - gx) + sl1(lm[tid*10+2*k+1] - gy);
        }
    } else {
        int bin = (int)(ce * BIN_SCALE);
        bin = bin < 0 ? 0 : (bin > NBINS - 1 ? NBINS - 1 : bin);
        atomicAdd(&shcnt[bin], 1u);
        atomicAdd(&shsum[bin], ce);
    }

    red[tid] = make_float4(lb, ll, pce_v, posf);
    __syncthreads();

    // flush per-image histogram
    for (int i = tid; i < NBINS; i += TPB) {
        if (shcnt[i]) atomicAdd(&hcnt[b * NBINS + i], shcnt[i]);
        if (shsum[i] != 0.0f) atomicAdd(&hsum[b * NBINS + i], shsum[i]);
    }

    // deterministic block tree-reduce of the 4 partial streams
    for (int s = TPB / 2; s; s >>= 1) {
        if (tid < s) {
            float4 o = red[tid + s];
            red[tid].x += o.x; red[tid].y += o.y; red[tid].z += o.z; red[tid].w += o.w;
        }
        __syncthreads();
    }
    if (tid == 0) {
        const int blk = b * gridDim.x + blockIdx.x;
        pbox[blk]   = red[0].x;
        plandm[blk] = red[0].y;
        pce[blk]    = red[0].z;
        pnpos[blk]  = red[0].w;
        const int cnt = (int)(red[0].w + 0.5f);
        if (cnt) atomicAdd(&npos_img[b], cnt);
    }
}

// WMMA-based sum of NB_TOTAL floats: A = ones(16x4), B = 64 data elements
// per tile, D = A*B + C chains the accumulator; 16 column sums folded by shfl.
__device__ __forceinline__ float wmma_sum(const float* __restrict__ arr, int lane) {
    v2f a; a[0] = 1.0f; a[1] = 1.0f;
    v8f c = {};
    for (int t = 0; t < NB_TOTAL / 64; ++t) {
        v2f bm;
        bm[0] = arr[t * 64 + lane];
        bm[1] = arr[t * 64 + 32 + lane];
        c = __builtin_amdgcn_wmma_f32_16x16x4_f32(false, a, false, bm,
                                                  (short)0, c, false, false);
    }
    float v = c[0];
    v += __shfl_xor(v, 1, 32);
    v += __shfl_xor(v, 2, 32);
    v += __shfl_xor(v, 4, 32);
    v += __shfl_xor(v, 8, 32);
    return v;   // total replicated across lanes (mod 16 groups)
}

// ---------------- kernel 3: finalize (single wave, EXEC all ones) ----------------
__global__ void finalize_kernel(const float* __restrict__ pbox,
                                const float* __restrict__ plandm,
                                const float* __restrict__ pce,
                                const float* __restrict__ pnpos,
                                const int* __restrict__ npos_img,
                                const unsigned int* __restrict__ hcnt,
                                const float* __restrict__ hsum,
                                float* __restrict__ out) {
    __shared__ float totals[4];
    __shared__ float negsum[BATCH];
    const int lane = threadIdx.x;

    float t0 = wmma_sum(pbox,   lane);
    float t1 = wmma_sum(plandm, lane);
    float t2 = wmma_sum(pce,    lane);
    float t3 = wmma_sum(pnpos,  lane);
    if (lane == 0) { totals[0] = t0; totals[1] = t1; totals[2] = t2; totals[3] = t3; }
    __syncthreads();

    // per-image hard-negative top-k sum via histogram scan
    if (lane < BATCH) {
        long long k = 7LL * (long long)npos_img[lane];
        if (k > NPRIORS - 1) k = NPRIORS - 1;
        long long acc = 0; float s = 0.0f;
        for (int bin = NBINS - 1; bin >= 0 && acc < k; --bin) {
            const unsigned int c = hcnt[lane * NBINS + bin];
            const float su = hsum[lane * NBINS + bin];
            if (acc + (long long)c <= k) { acc += c; s += su; }
            else {
                const long long rem = k - acc;
                s += (float)rem * (su / (float)c);
                acc = k;
            }
        }
        negsum[lane] = s;
    }
    __syncthreads();

    if (lane == 0) {
        const float N = fmaxf(totals[3], 1.0f);
        float neg = 0.0f;
        for (int i = 0; i < BATCH; ++i) neg += negsum[i];
        out[0] = totals[0] / N;            // loss_box
        out[1] = (totals[2] + neg) / N;    // loss_conf (pos CE + mined neg CE)
        out[2] = totals[1] / N;            // loss_landm
    }
}

extern "C" void kernel_launch(void* const* d_in, const int* in_sizes, int n_in,
                              void* d_out, int out_size, void* d_ws, size_t ws_size,
                              hipStream_t stream) {
    const float* conf    = (const float*)d_in[0];
    const float* loc     = (const float*)d_in[1];
    const float* landm   = (const float*)d_in[2];
    const float* priors  = (const float*)d_in[3];
    const float* targets = (const float*)d_in[4];
    float* out = (float*)d_out;

    char* ws = (char*)d_ws;
    unsigned long long* best = (unsigned long long*)(ws + OFF_BEST);
    int*          npos   = (int*)(ws + OFF_NPOS);
    unsigned int* hcnt   = (unsigned int*)(ws + OFF_HCNT);
    float*        hsum   = (float*)(ws + OFF_HSUM);
    float*        pbox   = (float*)(ws + OFF_PART + 0 * PART_STRIDE);
    float*        plandm = (float*)(ws + OFF_PART + 1 * PART_STRIDE);
    float*        pce    = (float*)(ws + OFF_PART + 2 * PART_STRIDE);
    float*        pnpos  = (float*)(ws + OFF_PART + 3 * PART_STRIDE);

    const int nzero = (OFF_PART - OFF_NPOS) / 4;   // npos + hcnt + hsum
    init_ws<<<(nzero + 255) / 256, 256, 0, stream>>>(best, (unsigned int*)(ws + OFF_NPOS), nzero);

    dim3 grid(BLK_P, BATCH);
    best_prior_kernel<<<grid, TPB, 0, stream>>>(priors, targets, best);
    main_kernel<<<grid, TPB, 0, stream>>>(conf, loc, landm, priors, targets, best,
                                          npos, hcnt, hsum, pbox, plandm, pce, pnpos);
    finalize_kernel<<<1, 32, 0, stream>>>(pbox, plandm, pce, pnpos, npos, hcnt, hsum, out);
}